// ScaledDotProductAttention_60902636257615
// MI455X (gfx1250) — compile-verified
//
#include <hip/hip_runtime.h>

typedef __attribute__((ext_vector_type(2))) float v2f;
typedef __attribute__((ext_vector_type(4))) float v4f;
typedef __attribute__((ext_vector_type(8))) float v8f;

#define LSEQ 2048
#define DDIM 64
#define NBH  64            // B*H = 4*16
#define SCALE_INV 0.125f   // 1/sqrt(64)
#define NEGVAL -1000000000.0f

// One wave (32 lanes) per block. Each wave owns a 16-row Q tile and the full
// 16 x 2048 fp32 score tile in LDS (128 KB; CDNA5 WGP has 320 KB).
extern "C" __global__ __launch_bounds__(32)
void sdpa_fwd_gfx1250(const float* __restrict__ qg,
                      const float* __restrict__ kg,
                      const float* __restrict__ vg,
                      const int*   __restrict__ mg,
                      float* __restrict__ outg,
                      float* __restrict__ attng)
{
    extern __shared__ float smem[];          // [16][LSEQ] fp32 scores -> probs

    const int lane = threadIdx.x;            // 0..31 (wave32)
    const int n16  = lane & 15;              // column / row-within-tile index
    const int hlf  = lane >> 4;              // 0: K(0,1)/rows 0-7, 1: K(2,3)/rows 8-15
    const int koff = hlf * 2;

    const int wg = blockIdx.x;               // 0..8191
    const int qt = wg & 127;                 // q tile within sequence
    const int bh = wg >> 7;                  // fused batch*head
    const int q0 = qt * 16;

    // ---- Load Q tile A-fragments (16 chunks of K=4), contiguous float2 per lane ----
    // A 16x4 fp32 layout: lanes 0-15 hold (K=0,K=1) of row=lane; lanes 16-31 hold (K=2,K=3).
    const float* qbase = qg + ((size_t)(bh * LSEQ + q0 + n16)) * DDIM + koff;
    v2f qa[16];
#pragma unroll
    for (int c = 0; c < 16; ++c)
        qa[c] = *(const v2f*)(qbase + 4 * c);

    const float* kbh  = kg + (size_t)bh * LSEQ * DDIM;
    const int*   mrow = mg + (size_t)bh * LSEQ;

    // ================= Pass 1: S = mask(QK^T / 8) -> LDS =================
    for (int kt = 0; kt < LSEQ / 16; ++kt) {
        // B-fragment of K^T (4x16): B[k][n] = K[kt*16+n][4c+k]; per lane a float2.
        const float* kb = kbh + (size_t)(kt * 16 + n16) * DDIM + koff;
        v8f acc = {};
#pragma unroll
        for (int c = 0; c < 16; ++c) {
            v2f b = *(const v2f*)(kb + 4 * c);
            acc = __builtin_amdgcn_wmma_f32_16x16x4_f32(
                      false, qa[c], false, b, (short)0, acc, false, false);
        }
        const int col = kt * 16 + n16;
        const int mk  = mrow[col];
#pragma unroll
        for (int j = 0; j < 8; ++j) {
            // C/D layout: VGPR j -> row j (lanes 0-15) or row j+8 (lanes 16-31), col = n16.
            float s = (mk != 0) ? acc[j] * SCALE_INV : NEGVAL;
            smem[(size_t)(j + hlf * 8) * LSEQ + col] = s;
        }
    }
    __syncthreads();

    // ====== Pass 2: exact softmax per row; stream attn out (NT); P stays in LDS ======
    float* attn_bh = attng + (size_t)(bh * LSEQ + q0) * LSEQ;
    v4f*   s4      = (v4f*)smem;
    for (int r = 0; r < 16; ++r) {
        v4f* row4 = s4 + (size_t)r * (LSEQ / 4);

        float mx = -3.0e38f;
        for (int c4 = lane; c4 < LSEQ / 4; c4 += 32) {
            v4f t = row4[c4];
            mx = fmaxf(mx, fmaxf(fmaxf(t.x, t.y), fmaxf(t.z, t.w)));
        }
#pragma unroll
        for (int off = 16; off > 0; off >>= 1)
            mx = fmaxf(mx, __shfl_xor(mx, off, 32));

        float sum = 0.0f;
        for (int c4 = lane; c4 < LSEQ / 4; c4 += 32) {
            v4f t = row4[c4];
            t.x = __expf(t.x - mx); t.y = __expf(t.y - mx);
            t.z = __expf(t.z - mx); t.w = __expf(t.w - mx);
            row4[c4] = t;
            sum += t.x + t.y + t.z + t.w;
        }
#pragma unroll
        for (int off = 16; off > 0; off >>= 1)
            sum += __shfl_xor(sum, off, 32);
        const float inv = 1.0f / sum;

        v4f* arow = (v4f*)(attn_bh + (size_t)r * LSEQ);
        for (int c4 = lane; c4 < LSEQ / 4; c4 += 32) {
            v4f t = row4[c4];
            t.x *= inv; t.y *= inv; t.z *= inv; t.w *= inv;
            row4[c4] = t;                          // normalized P back to LDS for PV
            __builtin_nontemporal_store(t, &arow[c4]);  // 1 GiB stream: don't pollute caches
        }
    }
    __syncthreads();

    // ================= Pass 3: O = P V (WMMA, 4 accumulators over D=64) =================
    const float* vbh  = vg + (size_t)bh * LSEQ * DDIM;
    const float* prow = smem + (size_t)n16 * LSEQ + koff;   // A-fragment row of P
    v8f o0 = {}, o1 = {}, o2 = {}, o3 = {};
    for (int kt = 0; kt < LSEQ / 4; ++kt) {
        v2f a = *(const v2f*)(prow + 4 * kt);               // P[m][4kt+koff .. +1] from LDS
        const float* vb = vbh + (size_t)(4 * kt + koff) * DDIM + n16;
        v2f b0 = { vb[0],      vb[DDIM] };
        o0 = __builtin_amdgcn_wmma_f32_16x16x4_f32(false, a, false, b0, (short)0, o0, false, false);
        v2f b1 = { vb[16],     vb[DDIM + 16] };
        o1 = __builtin_amdgcn_wmma_f32_16x16x4_f32(false, a, false, b1, (short)0, o1, false, false);
        v2f b2 = { vb[32],     vb[DDIM + 32] };
        o2 = __builtin_amdgcn_wmma_f32_16x16x4_f32(false, a, false, b2, (short)0, o2, false, false);
        v2f b3 = { vb[48],     vb[DDIM + 48] };
        o3 = __builtin_amdgcn_wmma_f32_16x16x4_f32(false, a, false, b3, (short)0, o3, false, false);
    }

    float* obase = outg + (size_t)(bh * LSEQ + q0) * DDIM;
#pragma unroll
    for (int j = 0; j < 8; ++j) {
        const size_t rbase = (size_t)(j + hlf * 8) * DDIM + n16;
        __builtin_nontemporal_store(o0[j], &obase[rbase +  0]);
        __builtin_nontemporal_store(o1[j], &obase[rbase + 16]);
        __builtin_nontemporal_store(o2[j], &obase[rbase + 32]);
        __builtin_nontemporal_store(o3[j], &obase[rbase + 48]);
    }
}

extern "C" void kernel_launch(void* const* d_in, const int* in_sizes, int n_in,
                              void* d_out, int out_size, void* d_ws, size_t ws_size,
                              hipStream_t stream) {
    (void)in_sizes; (void)n_in; (void)d_ws; (void)ws_size; (void)out_size;
    const float* q    = (const float*)d_in[0];
    const float* k    = (const float*)d_in[1];
    const float* v    = (const float*)d_in[2];
    const int*   mask = (const int*)d_in[3];

    float* out  = (float*)d_out;                       // [4,16,2048,64]
    float* attn = out + (size_t)NBH * LSEQ * DDIM;     // [4,16,2048,2048]

    const int    nblocks = NBH * (LSEQ / 16);          // 8192 waves
    const size_t lds     = (size_t)16 * LSEQ * sizeof(float);  // 128 KB / workgroup

    sdpa_fwd_gfx1250<<<dim3(nblocks), dim3(32), lds, stream>>>(q, k, v, mask, out, attn);
}